// PointerGenerator_54752243090123
// MI455X (gfx1250) — compile-verified
//
#include <hip/hip_runtime.h>
#include <hip/hip_bf16.h>

// ---------------- problem constants (fixed by reference) ----------------
#define BD     8
#define TDLEN  128
#define TELEN  512
#define DM     512           // d_model
#define DC     1024          // 2*d_model
#define VOC    50000
#define VEXT   50200         // VOC + 200 pad
#define ROWS   1024          // BD * TDLEN
#define NCHUNK 98            // ceil(VOC / 512)

typedef __attribute__((ext_vector_type(2))) float v2f;
typedef __attribute__((ext_vector_type(8))) float v8f;

// fp32 WMMA: D(16x16,f32) = A(16x4,f32) x B(4x16,f32) + C
// 8-arg pattern: (neg_a, A, neg_b, B, c_mod, C, reuse_a, reuse_b)
__device__ __forceinline__ v8f wmma4(v2f a, v2f b, v8f c) {
    return __builtin_amdgcn_wmma_f32_16x16x4_f32(false, a, false, b, (short)0, c,
                                                 false, false);
}

// ---------------------------------------------------------------------------
// Kernel A: result[1024,512] = ds @ Wd + ctx @ Wc
// grid = 64 blocks (one 16-row M tile each), 256 threads = 8 waves,
// wave w computes cols [w*64, w*64+64). A panel staged in LDS.
// ---------------------------------------------------------------------------
__global__ __launch_bounds__(256) void kA_result(const float* __restrict__ ds,
                                                 const float* __restrict__ ctx,
                                                 const float* __restrict__ Wd,
                                                 const float* __restrict__ Wc,
                                                 float* __restrict__ result) {
    __shared__ float ldsA[16 * 1028];  // big enough for 16 x 1024 (pad 4)

    const int m0   = blockIdx.x * 16;
    const int tid  = threadIdx.x;
    const int wv   = tid >> 5;
    const int lane = tid & 31;
    const int l16  = lane & 15;
    const int hk   = (lane >> 4) * 2;  // K sub-offset: 0 or 2
    const int n0   = wv * 64;

    v8f c[4] = {};

    // ---- phase 1: ds @ Wd, K = 512, LDS stride 516 (bank-conflict free) ----
    for (int i = tid; i < 16 * 512; i += 256) {
        int r = i >> 9, k = i & 511;
        ldsA[r * 516 + k] = ds[(m0 + r) * DM + k];
    }
    __syncthreads();
    for (int k = 0; k < 512; k += 4) {
        v2f a = *(const v2f*)&ldsA[l16 * 516 + k + hk];
        const float* w0 = Wd + (size_t)(k + hk) * DM;
#pragma unroll
        for (int t = 0; t < 4; ++t) {
            int n = n0 + t * 16 + l16;
            v2f b;
            b.x  = w0[n];
            b.y  = w0[DM + n];
            c[t] = wmma4(a, b, c[t]);
        }
    }
    __syncthreads();

    // ---- phase 2: ctx @ Wc, K = 1024, LDS stride 1028 ----
    for (int i = tid; i < 16 * 1024; i += 256) {
        int r = i >> 10, k = i & 1023;
        ldsA[r * 1028 + k] = ctx[(m0 + r) * DC + k];
    }
    __syncthreads();
    for (int k = 0; k < 1024; k += 4) {
        v2f a = *(const v2f*)&ldsA[l16 * 1028 + k + hk];
        const float* w0 = Wc + (size_t)(k + hk) * DM;
#pragma unroll
        for (int t = 0; t < 4; ++t) {
            int n = n0 + t * 16 + l16;
            v2f b;
            b.x  = w0[n];
            b.y  = w0[DM + n];
            c[t] = wmma4(a, b, c[t]);
        }
    }

    // ---- store: C/D layout: VGPR r -> row m0 + r + 8*(lane>=16), col n + l16
#pragma unroll
    for (int r = 0; r < 8; ++r) {
        int row = m0 + r + (lane >> 4) * 8;
#pragma unroll
        for (int t = 0; t < 4; ++t)
            result[(size_t)row * DM + n0 + t * 16 + l16] = c[t][r];
    }
}

// ---------------------------------------------------------------------------
// Gate: p_gen[row] = sigmoid(gen_w * (ds[row]·wdg + ctx[row]·wcg) + gen_b)
// one wave per row; 128 blocks x 256 threads = 1024 waves.
// ---------------------------------------------------------------------------
__global__ __launch_bounds__(256) void kGate(const float* __restrict__ ds,
                                             const float* __restrict__ ctx,
                                             const float* __restrict__ wdg,
                                             const float* __restrict__ wcg,
                                             const float* __restrict__ gw,
                                             const float* __restrict__ gb,
                                             float* __restrict__ p_gen) {
    int row  = (blockIdx.x * blockDim.x + threadIdx.x) >> 5;
    int lane = threadIdx.x & 31;
    if (row >= ROWS) return;
    float s = 0.f;
    for (int k = lane; k < DM; k += 32) s += ds[(size_t)row * DM + k] * wdg[k];
    for (int k = lane; k < DC; k += 32) s += ctx[(size_t)row * DC + k] * wcg[k];
#pragma unroll
    for (int m = 16; m; m >>= 1) s += __shfl_xor(s, m);
    if (lane == 0) {
        float g    = gw[0] * s + gb[0];
        p_gen[row] = 1.f / (1.f + __expf(-g));
    }
}

// ---------------------------------------------------------------------------
// Kernel B: logits = result @ Wg + bg  -> raw logits into d_out[:, :50000]
// plus per-(row, 512-col chunk) (max, sum_exp) partials.
// grid = (98, 32): block = 32 rows x 512 cols, 256 threads = 8 waves.
// Each wave: 2 M-tiles x 4 N-tiles -> 8 WMMAs per K-step sharing one set of
// B fragments (8 VMEM loads) => 2x arithmetic intensity on the Wg stream.
// ---------------------------------------------------------------------------
__global__ __launch_bounds__(256) void kLogits(const float* __restrict__ result,
                                               const float* __restrict__ Wg,
                                               const float* __restrict__ bg,
                                               float* __restrict__ out,
                                               float* __restrict__ pMax,
                                               float* __restrict__ pSum) {
    __shared__ float ldsA[32 * 516];   // 32-row A panel, pad 4 (bank stride 4)
    __shared__ float redM[8 * 32];
    __shared__ float redS[8 * 32];

    const int m0    = blockIdx.y * 32;
    const int chunk = blockIdx.x;
    const int tid   = threadIdx.x;
    const int wv    = tid >> 5;
    const int lane  = tid & 31;
    const int l16   = lane & 15;
    const int hk    = (lane >> 4) * 2;
    const int nbase = chunk * 512 + wv * 64;

    for (int i = tid; i < 32 * 512; i += 256) {
        int r = i >> 9, k = i & 511;
        ldsA[r * 516 + k] = result[(size_t)(m0 + r) * DM + k];
    }
    __syncthreads();

    int ncol[4], nclamp[4];
#pragma unroll
    for (int t = 0; t < 4; ++t) {
        ncol[t]   = nbase + t * 16 + l16;
        nclamp[t] = ncol[t] < VOC ? ncol[t] : (VOC - 1);
    }

    v8f c[2][4] = {};
    for (int k = 0; k < 512; k += 4) {
        v2f a0 = *(const v2f*)&ldsA[l16 * 516 + k + hk];
        v2f a1 = *(const v2f*)&ldsA[(16 + l16) * 516 + k + hk];
        const float* w0 = Wg + (size_t)(k + hk) * VOC;
        if (k + 16 < 512) {  // pull the B stream 16 K-rows ahead into WGP$
            __builtin_prefetch(w0 + (size_t)16 * VOC + nclamp[0], 0, 1);
            __builtin_prefetch(w0 + (size_t)16 * VOC + nclamp[2], 0, 1);
        }
#pragma unroll
        for (int t = 0; t < 4; ++t) {
            v2f b;
            b.x     = w0[nclamp[t]];
            b.y     = w0[VOC + nclamp[t]];
            c[0][t] = wmma4(a0, b, c[0][t]);
            c[1][t] = wmma4(a1, b, c[1][t]);
        }
    }

    float bgv[4];
#pragma unroll
    for (int t = 0; t < 4; ++t) bgv[t] = bg[nclamp[t]];

    const float NEG_INF = -__builtin_inff();
    float wmx[2][8], wsm[2][8];
#pragma unroll
    for (int m = 0; m < 2; ++m) {
#pragma unroll
        for (int r = 0; r < 8; ++r) {
            int   row = m0 + m * 16 + r + (lane >> 4) * 8;
            float x[4];
            float mx = NEG_INF;
#pragma unroll
            for (int t = 0; t < 4; ++t) {
                x[t] = c[m][t][r] + bgv[t];
                if (ncol[t] < VOC) {
                    out[(size_t)row * VEXT + ncol[t]] = x[t];
                    mx = fmaxf(mx, x[t]);
                }
            }
            // reduce over the 16-lane half-group (both row halves in parallel)
#pragma unroll
            for (int msk = 8; msk; msk >>= 1) mx = fmaxf(mx, __shfl_xor(mx, msk));
            float s = 0.f;
#pragma unroll
            for (int t = 0; t < 4; ++t)
                if (ncol[t] < VOC) s += __expf(x[t] - mx);
#pragma unroll
            for (int msk = 8; msk; msk >>= 1) s += __shfl_xor(s, msk);
            wmx[m][r] = mx;
            wsm[m][r] = s;
        }
    }
    if (l16 == 0) {
        int half = (lane >> 4) * 8;
#pragma unroll
        for (int m = 0; m < 2; ++m)
#pragma unroll
            for (int r = 0; r < 8; ++r) {
                redM[wv * 32 + m * 16 + half + r] = wmx[m][r];
                redS[wv * 32 + m * 16 + half + r] = wsm[m][r];
            }
    }
    __syncthreads();
    if (tid < 32) {
        float M = NEG_INF;
        for (int w = 0; w < 8; ++w) M = fmaxf(M, redM[w * 32 + tid]);
        float S = 0.f;
        for (int w = 0; w < 8; ++w) {
            float m = redM[w * 32 + tid];
            if (m > NEG_INF) S += redS[w * 32 + tid] * __expf(m - M);
        }
        pMax[(size_t)(m0 + tid) * NCHUNK + chunk] = M;
        pSum[(size_t)(m0 + tid) * NCHUNK + chunk] = S;
    }
}

// ---------------------------------------------------------------------------
// Kernel C: per-row log-sum-exp merge of the 98 chunk partials.
// ---------------------------------------------------------------------------
__global__ void kReduce(const float* __restrict__ pMax,
                        const float* __restrict__ pSum,
                        float* __restrict__ rowM, float* __restrict__ rowS) {
    int row = blockIdx.x * blockDim.x + threadIdx.x;
    if (row >= ROWS) return;
    const float NEG_INF = -__builtin_inff();
    float M = NEG_INF;
    for (int c = 0; c < NCHUNK; ++c) M = fmaxf(M, pMax[(size_t)row * NCHUNK + c]);
    float S = 0.f;
    for (int c = 0; c < NCHUNK; ++c) {
        float m = pMax[(size_t)row * NCHUNK + c];
        if (m > NEG_INF) S += pSum[(size_t)row * NCHUNK + c] * __expf(m - M);
    }
    rowM[row] = M;
    rowS[row] = S;
}

// ---------------------------------------------------------------------------
// Kernel D: out = exp(logit - M)/S * p_gen ; zero the 200-col pad. float4.
// ---------------------------------------------------------------------------
__global__ __launch_bounds__(256) void kNorm(float* __restrict__ out,
                                             const float* __restrict__ rowM,
                                             const float* __restrict__ rowS,
                                             const float* __restrict__ p_gen) {
    const int per_row4 = VEXT / 4;  // 12550 (VOC divisible by 4 too)
    long i = (long)blockIdx.x * blockDim.x + threadIdx.x;
    if (i >= (long)ROWS * per_row4) return;
    int row = (int)(i / per_row4);
    int c4  = (int)(i % per_row4) * 4;
    float4* p = (float4*)(out + (size_t)row * VEXT + c4);
    if (c4 < VOC) {
        float  M   = rowM[row];
        float  inv = p_gen[row] / rowS[row];
        float4 v   = *p;
        v.x = __expf(v.x - M) * inv;
        v.y = __expf(v.y - M) * inv;
        v.z = __expf(v.z - M) * inv;
        v.w = __expf(v.w - M) * inv;
        *p  = v;
    } else {
        *p = make_float4(0.f, 0.f, 0.f, 0.f);
    }
}

// ---------------------------------------------------------------------------
// Kernel E: scatter-add  out[row, idx[b,e]] += attn[b,t,e] * (1 - p_gen[row])
// ---------------------------------------------------------------------------
__global__ void kScatter(const float* __restrict__ attn,
                         const int* __restrict__ ext,
                         const float* __restrict__ p_gen,
                         float* __restrict__ out) {
    int i = blockIdx.x * blockDim.x + threadIdx.x;
    if (i >= BD * TDLEN * TELEN) return;
    int b   = i / (TDLEN * TELEN);
    int row = i / TELEN;  // b*TDLEN + t
    int e   = i % TELEN;
    int col = ext[b * TELEN + e];
    float v = attn[i] * (1.f - p_gen[row]);
    atomicAdd(out + (size_t)row * VEXT + col, v);
}

// ---------------------------------------------------------------------------
extern "C" void kernel_launch(void* const* d_in, const int* in_sizes, int n_in,
                              void* d_out, int out_size, void* d_ws, size_t ws_size,
                              hipStream_t stream) {
    (void)in_sizes; (void)n_in; (void)out_size; (void)ws_size;

    const float* attn = (const float*)d_in[0];   // [8,128,512]
    const float* ctx  = (const float*)d_in[1];   // [8,128,1024]
    const float* ds   = (const float*)d_in[2];   // [8,128,512]
    const int*   ext  = (const int*)d_in[3];     // [8,512]
    /* d_in[4] = extra_vocab scalar (compile-time constant here) */
    const float* Wd   = (const float*)d_in[5];   // [512,512]
    const float* Wc   = (const float*)d_in[6];   // [1024,512]
    const float* Wg   = (const float*)d_in[7];   // [512,50000]
    const float* bg   = (const float*)d_in[8];   // [50000]
    const float* wdg  = (const float*)d_in[9];   // [512,1]
    const float* wcg  = (const float*)d_in[10];  // [1024,1]
    const float* gw   = (const float*)d_in[11];  // scalar
    const float* gb   = (const float*)d_in[12];  // scalar

    float* out = (float*)d_out;

    // workspace layout (floats)
    float* ws     = (float*)d_ws;
    float* result = ws;                            // 1024*512
    float* p_gen  = result + ROWS * DM;            // 1024
    float* pMax   = p_gen + ROWS;                  // 1024*98
    float* pSum   = pMax + ROWS * NCHUNK;          // 1024*98
    float* rowM   = pSum + ROWS * NCHUNK;          // 1024
    float* rowS   = rowM + ROWS;                   // 1024

    // 1) result = ds@Wd + ctx@Wc
    kA_result<<<dim3(ROWS / 16), dim3(256), 0, stream>>>(ds, ctx, Wd, Wc, result);

    // 2) gate
    kGate<<<dim3(ROWS / 8), dim3(256), 0, stream>>>(ds, ctx, wdg, wcg, gw, gb, p_gen);

    // 3) big GEMM + bias -> raw logits in d_out, chunk softmax partials
    kLogits<<<dim3(NCHUNK, ROWS / 32), dim3(256), 0, stream>>>(result, Wg, bg, out,
                                                               pMax, pSum);

    // 4) per-row reduction of partials
    kReduce<<<dim3((ROWS + 255) / 256), dim3(256), 0, stream>>>(pMax, pSum, rowM, rowS);

    // 5) normalize + gate-scale + zero pad
    {
        long total4 = (long)ROWS * (VEXT / 4);
        int  blocks = (int)((total4 + 255) / 256);
        kNorm<<<dim3(blocks), dim3(256), 0, stream>>>(out, rowM, rowS, p_gen);
    }

    // 6) scatter-add attention mass
    {
        int total  = BD * TDLEN * TELEN;
        int blocks = (total + 255) / 256;
        kScatter<<<dim3(blocks), dim3(256), 0, stream>>>(attn, ext, p_gen, out);
    }
}